// MetaPathAggregator_80900003987573
// MI455X (gfx1250) — compile-verified
//
#include <hip/hip_runtime.h>

typedef __attribute__((ext_vector_type(2))) float v2f;
typedef __attribute__((ext_vector_type(4))) float f4;
typedef __attribute__((ext_vector_type(8))) float v8f;

#define NF 128           // IN_FEATS == HIDDEN
#define TM 32            // rows per block
#define LDSS 132         // LDS row stride (floats): 132 mod 64 = 4 banks/row shift -> conflict-free frags

__device__ __forceinline__ v8f wmma4(v2f a, v2f b, v8f c) {
    // D = A(16x4) x B(4x16) + C, full fp32
    return __builtin_amdgcn_wmma_f32_16x16x4_f32(
        /*neg_a=*/false, a, /*neg_b=*/false, b,
        /*c_mod=*/(short)0, c, /*reuse_a=*/false, /*reuse_b=*/false);
}

__global__ __launch_bounds__(64)
void metapath_wmma_kernel(const float* __restrict__ fmi,
                          const float* __restrict__ fg,
                          const float* __restrict__ fdr,
                          const float* __restrict__ Wdd,      // W_drug_disease (128x128)
                          const float* __restrict__ Wdisdr,   // W_disease_drug (128x128)
                          const float* __restrict__ Wdrug,    // (64x128)
                          const float* __restrict__ Wdis,     // (64x128)
                          const int*   __restrict__ mp,       // (rows,4)
                          float* __restrict__ out)            // (rows,128)
{
    __shared__ float sA[TM][LDSS];
    __shared__ float sR[TM][LDSS];
    __shared__ int   sIdx[TM * 4];

    const int t        = threadIdx.x;
    const int lane     = t & 31;
    const int wave     = t >> 5;           // 0..1
    const int rowbase  = wave * 16;        // this wave's 16-row M tile
    const int l15      = lane & 15;
    const int koff     = (lane >> 4) * 2;  // K offset within a 4-wide fragment
    const int blockRow = blockIdx.x * TM;  // first global row of this block

    // Stage the 32x4 index block into LDS (coalesced).
    sIdx[t]      = mp[blockRow * 4 + t];
    sIdx[t + 64] = mp[blockRow * 4 + 64 + t];
    __syncthreads();

    #pragma unroll 1
    for (int phase = 0; phase < 2; ++phase) {
        // phase 0: dis path  : A=(mi+g1)/2, R=g2/4+dr/2, Wmain=Wdd,    Wout=Wdis,  cols 64..127
        // phase 1: drug path : A=(dr+g2)/2, R=g1/4+mi/2, Wmain=Wdisdr, Wout=Wdrug, cols 0..63
        const float* tA0   = (phase == 0) ? fmi : fdr;
        const float* tA1   = fg;
        const float* tRq   = fg;
        const float* tRh   = (phase == 0) ? fdr : fmi;
        const int    cA0   = (phase == 0) ? 0 : 3;
        const int    cA1   = (phase == 0) ? 1 : 2;
        const int    cRq   = (phase == 0) ? 2 : 1;
        const int    cRh   = (phase == 0) ? 3 : 0;
        const float* Wmain = (phase == 0) ? Wdd : Wdisdr;
        const float* Wout  = (phase == 0) ? Wdis : Wdrug;
        const int    ncol0 = (phase == 0) ? 64 : 0;

        // ---- Stage A and R into LDS (float4 chunks; 32 rows x 32 chunks) ----
        #pragma unroll 4
        for (int i = t; i < TM * (NF / 4); i += 64) {
            const int r  = i >> 5;        // row within tile
            const int c4 = (i & 31) * 4;  // float4 column
            const int i0 = sIdx[r * 4 + cA0];
            const int i1 = sIdx[r * 4 + cA1];
            const int iq = sIdx[r * 4 + cRq];
            const int ih = sIdx[r * 4 + cRh];
            const f4 x = *(const f4*)(tA0 + i0 * NF + c4);
            const f4 y = *(const f4*)(tA1 + i1 * NF + c4);
            const f4 u = *(const f4*)(tRq + iq * NF + c4);
            const f4 v = *(const f4*)(tRh + ih * NF + c4);
            *(f4*)&sA[r][c4] = (x + y) * 0.5f;
            *(f4*)&sR[r][c4] = u * 0.25f + v * 0.5f;
        }
        __syncthreads();

        // ---- GEMM-1: T(16x128) = A(16x128) x Wmain^T, fp32 WMMA chain over K ----
        const v8f z = {0.f, 0.f, 0.f, 0.f, 0.f, 0.f, 0.f, 0.f};
        v8f acc[8];
        #pragma unroll
        for (int nt = 0; nt < 8; ++nt) acc[nt] = z;

        #pragma unroll 4
        for (int kk = 0; kk < 32; ++kk) {
            const int kb = kk * 4 + koff;
            const v2f a = *(const v2f*)&sA[rowbase + l15][kb];   // ds_load_b64
            #pragma unroll
            for (int nt = 0; nt < 8; ++nt) {
                // B[k][n] = Wmain[n][k] -> same float2 pattern from row-major W
                const v2f b = *(const v2f*)(Wmain + (nt * 16 + l15) * NF + kb);
                acc[nt] = wmma4(a, b, acc[nt]);
            }
        }

        // ---- feat = 0.25*T + R, written back into sA (wave-private rows) ----
        const int mb = rowbase + 8 * (lane >> 4);   // C-layout: vgpr v -> row mb+v
        #pragma unroll
        for (int nt = 0; nt < 8; ++nt) {
            const int col = nt * 16 + l15;
            #pragma unroll
            for (int v = 0; v < 8; ++v) {
                sA[mb + v][col] = 0.25f * acc[nt][v] + sR[mb + v][col];
            }
        }

        // ---- GEMM-2: out-half(16x64) = feat(16x128) x Wout^T ----
        v8f acc2[4];
        #pragma unroll
        for (int nt = 0; nt < 4; ++nt) acc2[nt] = z;

        #pragma unroll 4
        for (int kk = 0; kk < 32; ++kk) {
            const int kb = kk * 4 + koff;
            const v2f a = *(const v2f*)&sA[rowbase + l15][kb];
            #pragma unroll
            for (int nt = 0; nt < 4; ++nt) {
                const v2f b = *(const v2f*)(Wout + (nt * 16 + l15) * NF + kb);
                acc2[nt] = wmma4(a, b, acc2[nt]);
            }
        }

        // ---- Store this half of the output (streamed, nontemporal) ----
        const int grow = blockRow + mb;             // global row for vgpr v=0
        #pragma unroll
        for (int nt = 0; nt < 4; ++nt) {
            const int col = ncol0 + nt * 16 + l15;
            #pragma unroll
            for (int v = 0; v < 8; ++v) {
                __builtin_nontemporal_store(acc2[nt][v], out + (grow + v) * NF + col);
            }
        }
        __syncthreads();   // protect sA/sR before the next phase restages them
    }
}

extern "C" void kernel_launch(void* const* d_in, const int* in_sizes, int n_in,
                              void* d_out, int out_size, void* d_ws, size_t ws_size,
                              hipStream_t stream) {
    const float* fmi    = (const float*)d_in[0];
    const float* fg     = (const float*)d_in[1];
    const float* fdr    = (const float*)d_in[2];
    const float* Wdd    = (const float*)d_in[3];
    const float* Wdisdr = (const float*)d_in[4];
    const float* Wdrug  = (const float*)d_in[5];
    const float* Wdis   = (const float*)d_in[6];
    const int*   mp     = (const int*)d_in[7];
    float* out = (float*)d_out;

    const int n_rows = in_sizes[7] / 4;     // 1024 * 128 = 131072
    const int blocks = n_rows / TM;         // 4096

    metapath_wmma_kernel<<<dim3(blocks), dim3(64), 0, stream>>>(
        fmi, fg, fdr, Wdd, Wdisdr, Wdrug, Wdis, mp, out);
}